// RFM_14654428414716
// MI455X (gfx1250) — compile-verified
//
#include <hip/hip_runtime.h>
#include <math.h>

typedef __attribute__((ext_vector_type(2))) float v2f;
typedef __attribute__((ext_vector_type(8))) float v8f;

#define H  256
#define NV 512
#define NT 8192
#define V  64

// ---------------------------------------------------------------------------
// Kernel 1: Bv[v][h] = b1[h] + sum_i x_vehicle[current[v,0]][i] * W1[h][H+i]
// 64 blocks x 256 threads; vehicle row staged in LDS. ~4M FMA, negligible.
// ---------------------------------------------------------------------------
__global__ void k_vehicle_proj(const float* __restrict__ xv,
                               const int* __restrict__ current,
                               const float* __restrict__ W1,
                               const float* __restrict__ b1,
                               float* __restrict__ Bv) {
  __shared__ float veh[H];
  const int v = blockIdx.x;
  const int h = threadIdx.x;
  const int row = current[2 * v];           // current[:,0]
  veh[h] = xv[row * H + h];
  __syncthreads();
  const float* w = W1 + h * (2 * H) + H;    // W1[h, 256:512]
  float s = b1[h];
#pragma unroll 8
  for (int i = 0; i < H; ++i) s = fmaf(veh[i], w[i], s);
  Bv[v * H + h] = s;
}

// ---------------------------------------------------------------------------
// Kernel 2: A[t][h] = sum_i x_track[t][i] * W1[h][i]   (8192x256x256 f32 GEMM)
// One wave32 per 16x16 tile, V_WMMA_F32_16X16X4_F32, K-loop of 64.
// A-operand layout (16x4 f32): lane l in 0..15 -> row M=l; VGPR0 holds K=2*half,
// VGPR1 holds K=2*half+1  => the two K values are CONSECUTIVE addresses (b64).
// B-operand (4x16) mirrors it with N=l across lanes.
// C/D (16x16 f32, 8 VGPRs): row = r + 8*half, col = l.
// ---------------------------------------------------------------------------
__global__ void k_track_proj_wmma(const float* __restrict__ xt,
                                  const float* __restrict__ W1,
                                  float* __restrict__ A) {
  const int lane   = threadIdx.x & 31;
  const int wv     = threadIdx.x >> 5;
  const int tile   = blockIdx.x * 4 + wv;   // 8192 tiles = (8192/16)*(256/16)
  const int n_base = (tile & 15) * 16;      // h-tile
  const int m_base = (tile >> 4) * 16;      // t-tile
  const int half   = lane >> 4;
  const int l      = lane & 15;

  const float* aptr = xt + (m_base + l) * H + 2 * half;          // row-major X
  const float* bptr = W1 + (n_base + l) * (2 * H) + 2 * half;    // W1[h, 0:256]

  v8f c = {};
#pragma unroll 4
  for (int k = 0; k < H; k += 4) {
    v2f a = *(const v2f*)(aptr + k);   // X[m_base+l][k+2*half .. +1]
    v2f b = *(const v2f*)(bptr + k);   // W1[n_base+l][k+2*half .. +1]
    c = __builtin_amdgcn_wmma_f32_16x16x4_f32(
        /*neg_a=*/false, a, /*neg_b=*/false, b,
        /*c_mod=*/(short)0, c, /*reuse_a=*/false, /*reuse_b=*/false);
  }

  float* dptr = A + (m_base + 8 * half) * H + n_base + l;
#pragma unroll
  for (int r = 0; r < 8; ++r) dptr[r * H] = c[r];
}

// ---------------------------------------------------------------------------
// Kernel 3: fused epilogue.
//   scores[v,t] = relu(A[t]+Bv[v]) . W2[0] + b2[0]
//   pos[v,t]    = sigmoid( relu(A[t]+Bv[v]) . W2[1] + b2[1] )
// All 64 Bv rows in 64KB LDS, XOR-swizzled on low 6 bits of h so that the 32
// lanes of a wave (consecutive v, stride 256) hit 32 distinct banks.
// Each block: 256 threads cover v=0..63 x 4 t-lanes, looped over 4 t-groups
// (16 track rows per block -> 512 blocks), amortizing the LDS fill.
// ---------------------------------------------------------------------------
__global__ void k_pair_mlp(const float* __restrict__ A,
                           const float* __restrict__ Bv,
                           const float* __restrict__ W2,
                           const float* __restrict__ b2,
                           float* __restrict__ out) {
  __shared__ float BvL[V * H];              // exactly 64 KB
  const int tid = threadIdx.x;
#pragma unroll
  for (int j = 0; j < (V * H) / 256; ++j) {
    int idx = tid + j * 256;
    int v = idx >> 8, h = idx & 255;
    BvL[v * H + (h ^ v)] = Bv[idx];         // swizzled store
  }
  __syncthreads();

  const int v  = tid & 63;
  const int tl = tid >> 6;
  const float b2_0 = b2[0], b2_1 = b2[1];
  const float* bl = BvL + v * H;

  for (int j = 0; j < 4; ++j) {
    const int t = blockIdx.x * 16 + j * 4 + tl;
    const float* ar = A + t * H;            // wave-uniform-ish, L2 resident
    float s0 = 0.f, s1 = 0.f;
#pragma unroll 4
    for (int h = 0; h < H; ++h) {
      float x = ar[h] + bl[h ^ v];          // swizzled read, conflict-free
      x = fmaxf(x, 0.f);
      s0 = fmaf(x, W2[h], s0);              // W2[0,h] broadcast load
      s1 = fmaf(x, W2[H + h], s1);          // W2[1,h]
    }
    out[v * NT + t] = s0 + b2_0;            // scores, (64,8192) row-major
    const float z = s1 + b2_1;
    out[V * NT + v * NT + t] = 1.0f / (1.0f + __expf(-z));   // pos
  }
}

// ---------------------------------------------------------------------------
extern "C" void kernel_launch(void* const* d_in, const int* in_sizes, int n_in,
                              void* d_out, int out_size, void* d_ws, size_t ws_size,
                              hipStream_t stream) {
  const float* xv  = (const float*)d_in[0];   // x_vehicle (512,256)
  const float* xt  = (const float*)d_in[1];   // x_track   (8192,256)
  const int*   cur = (const int*)  d_in[2];   // current   (64,2)
  const float* W1  = (const float*)d_in[3];   // (256,512)
  const float* b1  = (const float*)d_in[4];   // (256,)
  const float* W2  = (const float*)d_in[5];   // (2,256)
  const float* b2  = (const float*)d_in[6];   // (2,)
  float* out = (float*)d_out;                 // scores(64*8192) ++ pos(64*8192)

  float* A  = (float*)d_ws;                   // 8192*256 f32 = 8 MB
  float* Bv = A + NT * H;                     // 64*256 f32

  k_vehicle_proj<<<V, H, 0, stream>>>(xv, cur, W1, b1, Bv);
  k_track_proj_wmma<<<(NT / 16) * (H / 16) / 4, 128, 0, stream>>>(xt, W1, A);
  k_pair_mlp<<<NT / 16, 256, 0, stream>>>(A, Bv, W2, b2, out);
}